// GATNet_32066225832230
// MI455X (gfx1250) — compile-verified
//
#include <hip/hip_runtime.h>
#include <math.h>

typedef __attribute__((ext_vector_type(2))) float v2f;
typedef __attribute__((ext_vector_type(8))) float v8f;

#define F_IN 128
#define HID 8
#define HEADS 8
#define C1 (HEADS * HID) /* 64 */
#define CLASSES 40
#define NEG_SLOPE 0.2f

// ---------- float <-> monotonically ordered uint (for atomic float max) ----------
__device__ __forceinline__ unsigned f2ord(float f) {
  unsigned u = __float_as_uint(f);
  return (u & 0x80000000u) ? ~u : (u | 0x80000000u);
}
__device__ __forceinline__ float ord2f(unsigned u) {
  return __uint_as_float((u & 0x80000000u) ? (u ^ 0x80000000u) : ~u);
}

__device__ __forceinline__ void edge_sd(const int* __restrict__ src,
                                        const int* __restrict__ dst,
                                        int E, int e, int& s, int& d) {
  if (e < E) { s = src[e]; d = dst[e]; } else { s = e - E; d = e - E; }
}

__device__ __forceinline__ float lrelu(float v) {
  return v > 0.0f ? v : NEG_SLOPE * v;
}

// =====================================================================
// GEMM1: H[n,64] = X[n,128] @ W[128,64], f32 WMMA 16x16x4
// block = 128 threads (4 waves); each wave owns a 16-col tile; 16 rows/block
// =====================================================================
__global__ __launch_bounds__(128) void gemm1_wmma(const float* __restrict__ X,
                                                  const float* __restrict__ W,
                                                  float* __restrict__ H, int n) {
  __shared__ float Xs[16][F_IN + 4];  // pad 4 -> rows 4 banks apart, conflict-free
  const int row0 = blockIdx.x * 16;
  const int tid = threadIdx.x;
  for (int i = tid; i < 16 * F_IN; i += 128) {
    int r = i >> 7, c = i & 127;
    int rr = row0 + r; if (rr >= n) rr = n - 1;
    Xs[r][c] = X[(size_t)rr * F_IN + c];
  }
  __syncthreads();

  const int wave = tid >> 5;   // 0..3 -> column tile
  const int lane = tid & 31;
  const int m = lane & 15;     // A row / B,C,D column within tile
  const int hi = lane >> 4;    // half-wave -> K pair (A/B), M+8 (C/D)
  const int col = wave * 16 + m;

  v8f acc = {0.f, 0.f, 0.f, 0.f, 0.f, 0.f, 0.f, 0.f};
  for (int k0 = 0; k0 < F_IN; k0 += 4) {
    const int k = k0 + hi * 2;
    v2f a, b;
    a[0] = Xs[m][k];
    a[1] = Xs[m][k + 1];
    b[0] = W[k * C1 + col];
    b[1] = W[(k + 1) * C1 + col];
    acc = __builtin_amdgcn_wmma_f32_16x16x4_f32(false, a, false, b,
                                                (short)0, acc, false, false);
  }
  const int mrow = row0 + hi * 8;
#pragma unroll
  for (int r = 0; r < 8; ++r) {
    if (mrow + r < n) H[(size_t)(mrow + r) * C1 + col] = acc[r];
  }
}

// =====================================================================
// GEMM2: G[n,40] = A[n,64] @ W[64,40], f32 WMMA 16x16x4 (cols padded to 48)
// block = 96 threads (3 waves)
// =====================================================================
__global__ __launch_bounds__(96) void gemm2_wmma(const float* __restrict__ X,
                                                 const float* __restrict__ W,
                                                 float* __restrict__ G, int n) {
  __shared__ float Xs[16][C1 + 4];
  const int row0 = blockIdx.x * 16;
  const int tid = threadIdx.x;
  for (int i = tid; i < 16 * C1; i += 96) {
    int r = i >> 6, c = i & 63;
    int rr = row0 + r; if (rr >= n) rr = n - 1;
    Xs[r][c] = X[(size_t)rr * C1 + c];
  }
  __syncthreads();

  const int wave = tid >> 5;
  const int lane = tid & 31;
  const int m = lane & 15;
  const int hi = lane >> 4;
  const int col = wave * 16 + m;
  const int colc = col < CLASSES ? col : CLASSES - 1;  // clamp: never OOB
  const float bmask = col < CLASSES ? 1.0f : 0.0f;

  v8f acc = {0.f, 0.f, 0.f, 0.f, 0.f, 0.f, 0.f, 0.f};
  for (int k0 = 0; k0 < C1; k0 += 4) {
    const int k = k0 + hi * 2;
    v2f a, b;
    a[0] = Xs[m][k];
    a[1] = Xs[m][k + 1];
    b[0] = W[k * CLASSES + colc] * bmask;
    b[1] = W[(k + 1) * CLASSES + colc] * bmask;
    acc = __builtin_amdgcn_wmma_f32_16x16x4_f32(false, a, false, b,
                                                (short)0, acc, false, false);
  }
  if (col < CLASSES) {
    const int mrow = row0 + hi * 8;
#pragma unroll
    for (int r = 0; r < 8; ++r) {
      if (mrow + r < n) G[(size_t)(mrow + r) * CLASSES + col] = acc[r];
    }
  }
}

// ---------- attention logits: al = sum_c h[.,h,c] * a[h,c] ----------
__global__ __launch_bounds__(256) void al1_kernel(const float* __restrict__ H,
                                                  const float* __restrict__ a_src,
                                                  const float* __restrict__ a_dst,
                                                  float* __restrict__ als,
                                                  float* __restrict__ ald, int n) {
  int t = blockIdx.x * blockDim.x + threadIdx.x;
  if (t >= n * HEADS) return;
  const int node = t >> 3, h = t & 7;
  const float* hp = H + (size_t)node * C1 + h * HID;
  const float* as = a_src + h * HID;
  const float* ad = a_dst + h * HID;
  float s = 0.f, d = 0.f;
#pragma unroll
  for (int c = 0; c < HID; ++c) { float v = hp[c]; s += v * as[c]; d += v * ad[c]; }
  als[t] = s; ald[t] = d;
}

__global__ __launch_bounds__(256) void al2_kernel(const float* __restrict__ G,
                                                  const float* __restrict__ a_src,
                                                  const float* __restrict__ a_dst,
                                                  float* __restrict__ als,
                                                  float* __restrict__ ald, int n) {
  int t = blockIdx.x * blockDim.x + threadIdx.x;
  if (t >= n) return;
  const float* g = G + (size_t)t * CLASSES;
  float s = 0.f, d = 0.f;
#pragma unroll
  for (int c = 0; c < CLASSES; ++c) { float v = g[c]; s += v * a_src[c]; d += v * a_dst[c]; }
  als[t] = s; ald[t] = d;
}

// ---------- layer-1 edge passes (thread per (edge, head)) ----------
__global__ __launch_bounds__(256) void edge_max1(const int* __restrict__ src,
                                                 const int* __restrict__ dst,
                                                 int E, int n,
                                                 const float* __restrict__ als,
                                                 const float* __restrict__ ald,
                                                 unsigned* __restrict__ m1) {
  int t = blockIdx.x * blockDim.x + threadIdx.x;
  if (t >= (E + n) * HEADS) return;
  const int e = t >> 3, h = t & 7;
  int s, d; edge_sd(src, dst, E, e, s, d);
  const float ev = lrelu(als[s * HEADS + h] + ald[d * HEADS + h]);
  atomicMax(&m1[d * HEADS + h], f2ord(ev));
}

__global__ __launch_bounds__(256) void edge_den1(const int* __restrict__ src,
                                                 const int* __restrict__ dst,
                                                 int E, int n,
                                                 const float* __restrict__ als,
                                                 const float* __restrict__ ald,
                                                 const unsigned* __restrict__ m1,
                                                 float* __restrict__ den1) {
  int t = blockIdx.x * blockDim.x + threadIdx.x;
  if (t >= (E + n) * HEADS) return;
  const int e = t >> 3, h = t & 7;
  int s, d; edge_sd(src, dst, E, e, s, d);
  const int di = d * HEADS + h;
  const float ev = lrelu(als[s * HEADS + h] + ald[di]);
  const float ex = expf(ev - ord2f(m1[di]));
  unsafeAtomicAdd(&den1[di], ex);
}

__global__ __launch_bounds__(256) void edge_acc1(const int* __restrict__ src,
                                                 const int* __restrict__ dst,
                                                 int E, int n,
                                                 const float* __restrict__ als,
                                                 const float* __restrict__ ald,
                                                 const unsigned* __restrict__ m1,
                                                 const float* __restrict__ den1,
                                                 const float* __restrict__ H,
                                                 float* __restrict__ acc) {
  int t = blockIdx.x * blockDim.x + threadIdx.x;
  if (t >= (E + n) * HEADS) return;
  const int e = t >> 3, h = t & 7;
  int s, d; edge_sd(src, dst, E, e, s, d);
  const int di = d * HEADS + h;
  const float ev = lrelu(als[s * HEADS + h] + ald[di]);
  const float alpha = expf(ev - ord2f(m1[di])) / den1[di];
  const float* hs = H + (size_t)s * C1 + h * HID;
  float* ap = acc + (size_t)d * C1 + h * HID;
#pragma unroll
  for (int c = 0; c < HID; ++c) unsafeAtomicAdd(&ap[c], hs[c] * alpha);
}

__global__ __launch_bounds__(256) void relu_bias1(float* __restrict__ acc,
                                                  const float* __restrict__ b1, int n) {
  int t = blockIdx.x * blockDim.x + threadIdx.x;
  if (t >= n * C1) return;
  acc[t] = fmaxf(acc[t] + b1[t & (C1 - 1)], 0.0f);
}

// ---------- layer-2 edge passes (single head, 40 classes) ----------
__global__ __launch_bounds__(256) void edge_max2(const int* __restrict__ src,
                                                 const int* __restrict__ dst,
                                                 int E, int n,
                                                 const float* __restrict__ als,
                                                 const float* __restrict__ ald,
                                                 unsigned* __restrict__ m2) {
  int t = blockIdx.x * blockDim.x + threadIdx.x;
  if (t >= E + n) return;
  int s, d; edge_sd(src, dst, E, t, s, d);
  atomicMax(&m2[d], f2ord(lrelu(als[s] + ald[d])));
}

__global__ __launch_bounds__(256) void edge_den2(const int* __restrict__ src,
                                                 const int* __restrict__ dst,
                                                 int E, int n,
                                                 const float* __restrict__ als,
                                                 const float* __restrict__ ald,
                                                 const unsigned* __restrict__ m2,
                                                 float* __restrict__ den2) {
  int t = blockIdx.x * blockDim.x + threadIdx.x;
  if (t >= E + n) return;
  int s, d; edge_sd(src, dst, E, t, s, d);
  const float ev = lrelu(als[s] + ald[d]);
  unsafeAtomicAdd(&den2[d], expf(ev - ord2f(m2[d])));
}

// thread per (edge, group-of-8-classes): 5 groups cover 40 classes
__global__ __launch_bounds__(256) void edge_acc2(const int* __restrict__ src,
                                                 const int* __restrict__ dst,
                                                 int E, int n,
                                                 const float* __restrict__ als,
                                                 const float* __restrict__ ald,
                                                 const unsigned* __restrict__ m2,
                                                 const float* __restrict__ den2,
                                                 const float* __restrict__ G,
                                                 float* __restrict__ acc) {
  int t = blockIdx.x * blockDim.x + threadIdx.x;
  if (t >= (E + n) * 5) return;
  const int e = t / 5, g = t - e * 5;
  int s, d; edge_sd(src, dst, E, e, s, d);
  const float ev = lrelu(als[s] + ald[d]);
  const float alpha = expf(ev - ord2f(m2[d])) / den2[d];
  const float* gs = G + (size_t)s * CLASSES + g * 8;
  float* ap = acc + (size_t)d * CLASSES + g * 8;
#pragma unroll
  for (int c = 0; c < 8; ++c) unsafeAtomicAdd(&ap[c], gs[c] * alpha);
}

// ---------- final: add bias + log_softmax over 40 classes ----------
__global__ __launch_bounds__(256) void logsoftmax_kernel(const float* __restrict__ acc,
                                                         const float* __restrict__ b2,
                                                         float* __restrict__ out, int n) {
  int t = blockIdx.x * blockDim.x + threadIdx.x;
  if (t >= n) return;
  const float* row = acc + (size_t)t * CLASSES;
  float v[CLASSES];
  float mx = -3.4e38f;
#pragma unroll
  for (int c = 0; c < CLASSES; ++c) { v[c] = row[c] + b2[c]; mx = fmaxf(mx, v[c]); }
  float ssum = 0.f;
#pragma unroll
  for (int c = 0; c < CLASSES; ++c) ssum += expf(v[c] - mx);
  const float lse = logf(ssum);
  float* o = out + (size_t)t * CLASSES;
#pragma unroll
  for (int c = 0; c < CLASSES; ++c) o[c] = v[c] - mx - lse;
}

// =====================================================================
extern "C" void kernel_launch(void* const* d_in, const int* in_sizes, int n_in,
                              void* d_out, int out_size, void* d_ws, size_t ws_size,
                              hipStream_t stream) {
  const float* x   = (const float*)d_in[0];
  const int*   ei  = (const int*)d_in[1];
  const float* W1  = (const float*)d_in[2];
  const float* as1 = (const float*)d_in[3];
  const float* ad1 = (const float*)d_in[4];
  const float* b1  = (const float*)d_in[5];
  const float* W2  = (const float*)d_in[6];
  const float* as2 = (const float*)d_in[7];
  const float* ad2 = (const float*)d_in[8];
  const float* b2  = (const float*)d_in[9];
  float* out = (float*)d_out;

  const int n = in_sizes[0] / F_IN;
  const int E = in_sizes[1] / 2;
  const int* src = ei;       // edge_index[0]
  const int* dst = ei + E;   // edge_index[1]
  const int Etot = E + n;    // + self loops

  // ---- workspace layout ----
  float* p = (float*)d_ws;
  float* h1    = p; p += (size_t)n * C1;
  float* al_s1 = p; p += (size_t)n * HEADS;
  float* al_d1 = p; p += (size_t)n * HEADS;
  float* g2    = p; p += (size_t)n * CLASSES;
  float* al_s2 = p; p += n;
  float* al_d2 = p; p += n;
  char* accbase = (char*)p;                      // zeroed every call
  unsigned* m1   = (unsigned*)p; p += (size_t)n * HEADS;
  float*    den1 = p;            p += (size_t)n * HEADS;
  float*    acc1 = p;            p += (size_t)n * C1;
  unsigned* m2   = (unsigned*)p; p += n;
  float*    den2 = p;            p += n;
  float*    acc2 = p;            p += (size_t)n * CLASSES;
  const size_t accBytes = (char*)p - accbase;
  hipMemsetAsync(accbase, 0, accBytes, stream);  // ord(0) == -inf sentinel for m1/m2

  const int B = 256;
  const int rowTiles = (n + 15) / 16;

  // ---- layer 1 ----
  gemm1_wmma<<<rowTiles, 128, 0, stream>>>(x, W1, h1, n);
  al1_kernel<<<(n * HEADS + B - 1) / B, B, 0, stream>>>(h1, as1, ad1, al_s1, al_d1, n);
  edge_max1<<<(Etot * HEADS + B - 1) / B, B, 0, stream>>>(src, dst, E, n, al_s1, al_d1, m1);
  edge_den1<<<(Etot * HEADS + B - 1) / B, B, 0, stream>>>(src, dst, E, n, al_s1, al_d1, m1, den1);
  edge_acc1<<<(Etot * HEADS + B - 1) / B, B, 0, stream>>>(src, dst, E, n, al_s1, al_d1, m1, den1, h1, acc1);
  relu_bias1<<<(n * C1 + B - 1) / B, B, 0, stream>>>(acc1, b1, n);

  // ---- layer 2 ----
  gemm2_wmma<<<rowTiles, 96, 0, stream>>>(acc1, W2, g2, n);
  al2_kernel<<<(n + B - 1) / B, B, 0, stream>>>(g2, as2, ad2, al_s2, al_d2, n);
  edge_max2<<<(Etot + B - 1) / B, B, 0, stream>>>(src, dst, E, n, al_s2, al_d2, m2);
  edge_den2<<<(Etot + B - 1) / B, B, 0, stream>>>(src, dst, E, n, al_s2, al_d2, m2, den2);
  edge_acc2<<<(Etot * 5 + B - 1) / B, B, 0, stream>>>(src, dst, E, n, al_s2, al_d2, m2, den2, g2, acc2);
  logsoftmax_kernel<<<(n + B - 1) / B, B, 0, stream>>>(acc2, b2, out, n);
}